// GNN_only_58506044506790
// MI455X (gfx1250) — compile-verified
//
#include <hip/hip_runtime.h>
#include <cstddef>

namespace {

constexpr int kN = 50000;   // nodes
constexpr int kE = 800000;  // edges
constexpr int kC = 64;      // channels (IN_C == HID == OUT_C)
constexpr int kT = 16;      // time steps

typedef __attribute__((ext_vector_type(2))) float v2f;
typedef __attribute__((ext_vector_type(8))) float v8f;

// ---------------------------------------------------------------------------
// xa[n][c] = mean_t x[n][c][t]   (x is (N, C, T) row-major, T=16 contiguous)
// ---------------------------------------------------------------------------
__global__ __launch_bounds__(256) void mean_t_kernel(const float* __restrict__ x,
                                                     float* __restrict__ xa) {
  int i = blockIdx.x * blockDim.x + threadIdx.x;  // over N*C
  if (i >= kN * kC) return;
  const float4* p = reinterpret_cast<const float4*>(x + (size_t)i * kT);
  float4 a = p[0], b = p[1], c = p[2], d = p[3];
  float s = (((a.x + a.y) + (a.z + a.w)) + ((b.x + b.y) + (b.z + b.w))) +
            (((c.x + c.y) + (c.z + c.w)) + ((d.x + d.y) + (d.z + d.w)));
  xa[i] = s * (1.0f / 16.0f);
}

// deg[n] starts at 1.0 (reference adds +1.0 after the segment sum)
__global__ __launch_bounds__(256) void deg_init_kernel(float* __restrict__ deg) {
  int i = blockIdx.x * blockDim.x + threadIdx.x;
  if (i < kN) deg[i] = 1.0f;
}

// ew[e] = edge_attr[e][T-1];  deg[dst[e]] += ew[e]
__global__ __launch_bounds__(256) void deg_edge_kernel(const float* __restrict__ edge_attr,
                                                       const int* __restrict__ dst,
                                                       float* __restrict__ ew,
                                                       float* __restrict__ deg) {
  int e = blockIdx.x * blockDim.x + threadIdx.x;
  if (e >= kE) return;
  float w = edge_attr[(size_t)e * kT + (kT - 1)];
  ew[e] = w;
  __hip_atomic_fetch_add(&deg[dst[e]], w, __ATOMIC_RELAXED, __HIP_MEMORY_SCOPE_AGENT);
}

__global__ __launch_bounds__(256) void dinv_kernel(const float* __restrict__ deg,
                                                   float* __restrict__ dinv) {
  int i = blockIdx.x * blockDim.x + threadIdx.x;
  if (i >= kN) return;
  float d = deg[i];
  dinv[i] = (d > 0.0f) ? rsqrtf(d) : 0.0f;
}

// ---------------------------------------------------------------------------
// Out[M x 64] = A[M x 64] @ W[64 x 64] (+ bias) using V_WMMA_F32_16X16X4_F32.
// One wave per 16-row tile; 4 col-tiles of 16 held in 4 v8f accumulators.
// A frag (16x4 f32): lanes 0-15 K={k0,k0+1}, lanes 16-31 K={k0+2,k0+3}.
// B frag (4x16 f32): VGPR0 = rows K=k0 / k0+2, VGPR1 = rows K=k0+1 / k0+3.
// C/D (16x16 f32): VGPR j -> row j (lanes 0-15) / row j+8 (lanes 16-31).
// ---------------------------------------------------------------------------
__global__ __launch_bounds__(128) void gemm64_wmma_kernel(const float* __restrict__ A,
                                                          const float* __restrict__ W,
                                                          const float* __restrict__ bias,
                                                          float* __restrict__ Out) {
  const int lane = threadIdx.x & 31;
  const int tile = blockIdx.x * 4 + (threadIdx.x >> 5);  // 16-row tile index
  if (tile * 16 >= kN) return;                           // uniform per wave
  const int row0 = tile * 16;
  const int l15 = lane & 15;
  const int kh = (lane >> 4) << 1;  // 0 (lanes 0-15) or 2 (lanes 16-31)

  v8f acc0 = {}, acc1 = {}, acc2 = {}, acc3 = {};

#if __has_builtin(__builtin_amdgcn_wmma_f32_16x16x4_f32)
  const float* __restrict__ Arow = A + (size_t)(row0 + l15) * kC;
  for (int k0 = 0; k0 < kC; k0 += 4) {
    v2f a;
    a.x = Arow[k0 + kh];
    a.y = Arow[k0 + kh + 1];
    const float* __restrict__ Wk = W + (size_t)(k0 + kh) * kC;
    v2f b0, b1, b2, b3;
    b0.x = Wk[l15 + 0];  b0.y = Wk[kC + l15 + 0];
    b1.x = Wk[l15 + 16]; b1.y = Wk[kC + l15 + 16];
    b2.x = Wk[l15 + 32]; b2.y = Wk[kC + l15 + 32];
    b3.x = Wk[l15 + 48]; b3.y = Wk[kC + l15 + 48];
    acc0 = __builtin_amdgcn_wmma_f32_16x16x4_f32(false, a, false, b0, (short)0, acc0, false, false);
    acc1 = __builtin_amdgcn_wmma_f32_16x16x4_f32(false, a, false, b1, (short)0, acc1, false, false);
    acc2 = __builtin_amdgcn_wmma_f32_16x16x4_f32(false, a, false, b2, (short)0, acc2, false, false);
    acc3 = __builtin_amdgcn_wmma_f32_16x16x4_f32(false, a, false, b3, (short)0, acc3, false, false);
  }

  float bv0 = 0.f, bv1 = 0.f, bv2 = 0.f, bv3 = 0.f;
  if (bias) {
    bv0 = bias[l15 + 0];
    bv1 = bias[l15 + 16];
    bv2 = bias[l15 + 32];
    bv3 = bias[l15 + 48];
  }
  const int rsel = (lane >> 4) << 3;  // rows 0-7 (lanes 0-15) or 8-15
  for (int j = 0; j < 8; ++j) {
    float* __restrict__ orow = Out + (size_t)(row0 + rsel + j) * kC;
    orow[l15 + 0]  = acc0[j] + bv0;
    orow[l15 + 16] = acc1[j] + bv1;
    orow[l15 + 32] = acc2[j] + bv2;
    orow[l15 + 48] = acc3[j] + bv3;
  }
#else
  // Scalar fallback (should not be taken on gfx1250)
  for (int idx = lane; idx < 16 * kC; idx += 32) {
    int r = idx >> 6, c = idx & 63;
    float s = 0.f;
    for (int k = 0; k < kC; ++k) s += A[(size_t)(row0 + r) * kC + k] * W[(size_t)k * kC + c];
    Out[(size_t)(row0 + r) * kC + c] = s + (bias ? bias[c] : 0.f);
  }
  (void)acc0; (void)acc1; (void)acc2; (void)acc3; (void)kh;
#endif
}

// ---------------------------------------------------------------------------
// agg[dst[e]][c] += h[src[e]][c] * (dinv[src]*ew[e]*dinv[dst])
// 16 threads / edge, float4 gather + 4 scalar fp32 atomics each.
// h (12.8 MB) is L2-resident on MI455X (192 MB L2).
// ---------------------------------------------------------------------------
__global__ __launch_bounds__(256) void scatter_kernel(const float* __restrict__ h,
                                                      const int* __restrict__ src,
                                                      const int* __restrict__ dst,
                                                      const float* __restrict__ ew,
                                                      const float* __restrict__ dinv,
                                                      float* __restrict__ agg) {
  int t = blockIdx.x * blockDim.x + threadIdx.x;
  int e = t >> 4;
  if (e >= kE) return;
  int c4 = (t & 15) << 2;
  int s = src[e];
  int d = dst[e];
  float norm = dinv[s] * ew[e] * dinv[d];
  float4 hv = *reinterpret_cast<const float4*>(h + (size_t)s * kC + c4);
  float* base = agg + (size_t)d * kC + c4;
  __hip_atomic_fetch_add(base + 0, hv.x * norm, __ATOMIC_RELAXED, __HIP_MEMORY_SCOPE_AGENT);
  __hip_atomic_fetch_add(base + 1, hv.y * norm, __ATOMIC_RELAXED, __HIP_MEMORY_SCOPE_AGENT);
  __hip_atomic_fetch_add(base + 2, hv.z * norm, __ATOMIC_RELAXED, __HIP_MEMORY_SCOPE_AGENT);
  __hip_atomic_fetch_add(base + 3, hv.w * norm, __ATOMIC_RELAXED, __HIP_MEMORY_SCOPE_AGENT);
}

// agg[n][c] = maybe_relu(agg[n][c] + h[n][c]*dinv[n]^2 + bias[c])   (in place)
__global__ __launch_bounds__(256) void post_kernel(float* __restrict__ agg,
                                                   const float* __restrict__ h,
                                                   const float* __restrict__ dinv,
                                                   const float* __restrict__ bias,
                                                   int relu) {
  int i = blockIdx.x * blockDim.x + threadIdx.x;
  if (i >= kN * kC) return;
  int n = i >> 6;
  float di = dinv[n];
  float v = agg[i] + h[i] * (di * di) + bias[i & 63];
  if (relu) v = fmaxf(v, 0.0f);
  agg[i] = v;
}

}  // namespace

extern "C" void kernel_launch(void* const* d_in, const int* in_sizes, int n_in,
                              void* d_out, int out_size, void* d_ws, size_t ws_size,
                              hipStream_t stream) {
  (void)in_sizes; (void)n_in; (void)out_size; (void)ws_size;

  const float* x    = (const float*)d_in[0];  // (N, C, T)
  const int*   ei   = (const int*)d_in[1];    // (2, E)
  const float* ea   = (const float*)d_in[2];  // (E, T)
  const float* W1   = (const float*)d_in[3];
  const float* b1   = (const float*)d_in[4];
  const float* W2   = (const float*)d_in[5];
  const float* b2   = (const float*)d_in[6];
  const float* Wout = (const float*)d_in[7];
  const float* bout = (const float*)d_in[8];
  float*       out  = (float*)d_out;          // (N, C)

  const int* src = ei;
  const int* dst = ei + kE;

  // Workspace layout (floats): xa | h | agg | deg | dinv | ew  (~42 MB)
  float* ws   = (float*)d_ws;
  float* xa   = ws;
  float* h    = xa  + (size_t)kN * kC;
  float* agg  = h   + (size_t)kN * kC;
  float* deg  = agg + (size_t)kN * kC;
  float* dinv = deg + kN;
  float* ew   = dinv + kN;

  const int nc_blocks   = (kN * kC + 255) / 256;
  const int n_blocks    = (kN + 255) / 256;
  const int e_blocks    = (kE + 255) / 256;
  const int sc_blocks   = (kE * 16 + 255) / 256;
  const int gemm_blocks = (((kN + 15) / 16) + 3) / 4;  // 3125 tiles / 4 waves

  // Prologue: temporal mean, degree, dinv
  mean_t_kernel<<<nc_blocks, 256, 0, stream>>>(x, xa);
  deg_init_kernel<<<n_blocks, 256, 0, stream>>>(deg);
  deg_edge_kernel<<<e_blocks, 256, 0, stream>>>(ea, dst, ew, deg);
  dinv_kernel<<<n_blocks, 256, 0, stream>>>(deg, dinv);

  // GCN layer 1: h = xa @ W1 ; agg = scatter ; agg = relu(agg + h*dinv^2 + b1)
  gemm64_wmma_kernel<<<gemm_blocks, 128, 0, stream>>>(xa, W1, nullptr, h);
  hipMemsetAsync(agg, 0, (size_t)kN * kC * sizeof(float), stream);
  scatter_kernel<<<sc_blocks, 256, 0, stream>>>(h, src, dst, ew, dinv, agg);
  post_kernel<<<nc_blocks, 256, 0, stream>>>(agg, h, dinv, b1, 1);

  // GCN layer 2: h = agg @ W2 ; xa reused as agg2 ; xa = agg2 + h*dinv^2 + b2
  gemm64_wmma_kernel<<<gemm_blocks, 128, 0, stream>>>(agg, W2, nullptr, h);
  hipMemsetAsync(xa, 0, (size_t)kN * kC * sizeof(float), stream);
  scatter_kernel<<<sc_blocks, 256, 0, stream>>>(h, src, dst, ew, dinv, xa);
  post_kernel<<<nc_blocks, 256, 0, stream>>>(xa, h, dinv, b2, 0);

  // Output head: out = xa @ Wout + bout
  gemm64_wmma_kernel<<<gemm_blocks, 128, 0, stream>>>(xa, Wout, bout, out);
}